// LightGCN_91259465105579
// MI455X (gfx1250) — compile-verified
//
#include <hip/hip_runtime.h>

#define EMB_DIM 128

typedef __attribute__((ext_vector_type(2))) float v2f;
typedef __attribute__((ext_vector_type(8))) float v8f;

// ---------------------------------------------------------------------------
// Kernel 1: build all_emb = concat(user_emb, item_emb) into both acc and h0.
// One thread per float4 (32 float4 per node row).
// ---------------------------------------------------------------------------
__global__ void lgcn_init_kernel(const float4* __restrict__ ue,
                                 const float4* __restrict__ ie,
                                 float4* __restrict__ acc,
                                 float4* __restrict__ h,
                                 int user4, int total4) {
    int i = blockIdx.x * blockDim.x + threadIdx.x;
    if (i >= total4) return;
    float4 v = (i < user4) ? ue[i] : ie[i - user4];
    acc[i] = v;
    h[i]   = v;
}

// ---------------------------------------------------------------------------
// Kernel 2: SpMM scatter. One wave (32 lanes) per edge; lane handles 4 dims.
// Gather x[col] as coalesced float4 (512B/wave), scale by val, scatter with
// global_atomic_add_f32 to y[row]. Edge arrays are streamed once per layer ->
// non-temporal loads so they don't evict the L2-resident node matrix.
// ---------------------------------------------------------------------------
__global__ void lgcn_spmm_kernel(const int*   __restrict__ row,
                                 const int*   __restrict__ col,
                                 const float* __restrict__ val,
                                 const float* __restrict__ x,
                                 float*       __restrict__ y,
                                 int n_edges) {
    unsigned gid  = blockIdx.x * blockDim.x + threadIdx.x;
    unsigned edge = gid >> 5;
    unsigned lane = gid & 31u;
    if (edge >= (unsigned)n_edges) return;

    int   r = __builtin_nontemporal_load(&row[edge]);
    int   c = __builtin_nontemporal_load(&col[edge]);
    float v = __builtin_nontemporal_load(&val[edge]);

    const float4* xs = (const float4*)(x + (size_t)c * EMB_DIM);
    float4 xv = xs[lane];

    float* yp = y + (size_t)r * EMB_DIM + lane * 4;
    atomicAdd(yp + 0, v * xv.x);
    atomicAdd(yp + 1, v * xv.y);
    atomicAdd(yp + 2, v * xv.z);
    atomicAdd(yp + 3, v * xv.w);
}

// ---------------------------------------------------------------------------
// Kernel 3: acc += h  (one float4 per thread)
// ---------------------------------------------------------------------------
__global__ void lgcn_accum_kernel(float4* __restrict__ acc,
                                  const float4* __restrict__ h,
                                  int total4) {
    int i = blockIdx.x * blockDim.x + threadIdx.x;
    if (i >= total4) return;
    float4 a = acc[i];
    float4 b = h[i];
    a.x += b.x; a.y += b.y; a.z += b.z; a.w += b.w;
    acc[i] = a;
}

// ---------------------------------------------------------------------------
// Kernel 4: batched diagonal dot via V_WMMA_F32_16X16X4_F32.
// One wave per 16 batch entries. A = 16 user rows (16x4 tile), B = 16 item
// rows as columns (4x16 tile), D accumulates 16x16 scores over K=128 in
// steps of 4; we keep only the diagonal.
//   A layout (16x4 f32): lanes 0-15 = M, VGPR0 holds K0 (lo half)/K2 (hi
//   half), VGPR1 holds K1/K3  -> each lane loads float2 at k + 2*(lane>=16).
//   B layout mirrors with N across lanes -> same float2 addressing on items.
//   D diag: m=0..7 -> lane m, VGPR m; m=8..15 -> lane m+16, VGPR m-8.
// ---------------------------------------------------------------------------
__global__ void lgcn_score_wmma_kernel(const float* __restrict__ acc,
                                       const int*   __restrict__ users,
                                       const int*   __restrict__ items,
                                       float*       __restrict__ out,
                                       int n_users, int batch, float scale) {
    unsigned gid  = blockIdx.x * blockDim.x + threadIdx.x;
    unsigned tile = gid >> 5;
    unsigned lane = gid & 31u;
    unsigned base = tile * 16;
    if (base >= (unsigned)batch) return;   // never taken with exact grid

    unsigned m  = lane & 15u;      // row (user) / col (item) within tile
    unsigned hi = lane >> 4;       // 0 -> K pair {0,1}, 1 -> K pair {2,3}

    int unode = users[base + m];
    int inode = items[base + m] + n_users;
    const float* up = acc + (size_t)unode * EMB_DIM + hi * 2;
    const float* ip = acc + (size_t)inode * EMB_DIM + hi * 2;

    v8f c = {};
#pragma unroll
    for (int k = 0; k < EMB_DIM; k += 4) {
        float2 af = *(const float2*)(up + k);
        float2 bf = *(const float2*)(ip + k);
        v2f a; a.x = af.x; a.y = af.y;
        v2f b; b.x = bf.x; b.y = bf.y;
        // 8 args: (neg_a, A, neg_b, B, c_mod, C, reuse_a, reuse_b)
        c = __builtin_amdgcn_wmma_f32_16x16x4_f32(
                false, a, false, b, (short)0, c, false, false);
    }

    float d = 0.0f;
    int   om = -1;
    if (lane < 8u) {                    // diag m = 0..7
        d  = c[lane];
        om = (int)(base + lane);
    } else if (lane >= 24u) {           // diag m = 8..15 (lane = m + 16)
        d  = c[lane - 24u];
        om = (int)(base + lane - 16u);
    }
    if (om >= 0) out[om] = d * scale;
}

// ---------------------------------------------------------------------------
// Host launcher
// ---------------------------------------------------------------------------
extern "C" void kernel_launch(void* const* d_in, const int* in_sizes, int n_in,
                              void* d_out, int out_size, void* d_ws, size_t ws_size,
                              hipStream_t stream) {
    (void)n_in; (void)out_size; (void)ws_size;

    const float* user_emb = (const float*)d_in[0];
    const float* item_emb = (const float*)d_in[1];
    const float* adj_val  = (const float*)d_in[2];
    const int*   users    = (const int*)  d_in[3];
    const int*   items    = (const int*)  d_in[4];
    const int*   adj_row  = (const int*)  d_in[5];
    const int*   adj_col  = (const int*)  d_in[6];
    // d_in[7] = n_layers (device scalar). Host cannot read device memory under
    // graph capture; harness setup fixes it to 3.
    const int n_layers = 3;

    const int n_users = in_sizes[0] / EMB_DIM;     // 100000
    const int n_items = in_sizes[1] / EMB_DIM;     // 50000
    const int n_nodes = n_users + n_items;         // 150000
    const int n_edges = in_sizes[5];               // 4800000
    const int batch   = in_sizes[3];               // 16384

    float* out = (float*)d_out;

    const size_t node_floats = (size_t)n_nodes * EMB_DIM;  // 19.2M
    float* acc  = (float*)d_ws;
    float* bufA = acc  + node_floats;
    float* bufB = bufA + node_floats;

    const int total4 = n_nodes * (EMB_DIM / 4);            // 4.8M float4
    const int user4  = n_users * (EMB_DIM / 4);

    // acc = h0 = concat(user_emb, item_emb)
    lgcn_init_kernel<<<(total4 + 255) / 256, 256, 0, stream>>>(
        (const float4*)user_emb, (const float4*)item_emb,
        (float4*)acc, (float4*)bufA, user4, total4);

    // 3 propagation layers: h <- A h ; acc += h   (ping-pong bufA/bufB)
    float* h  = bufA;
    float* hn = bufB;
    const long long spmm_threads = (long long)n_edges * 32;
    const int spmm_blocks = (int)((spmm_threads + 255) / 256);
    for (int l = 0; l < n_layers; ++l) {
        hipMemsetAsync(hn, 0, node_floats * sizeof(float), stream);
        lgcn_spmm_kernel<<<spmm_blocks, 256, 0, stream>>>(
            adj_row, adj_col, adj_val, h, hn, n_edges);
        lgcn_accum_kernel<<<(total4 + 255) / 256, 256, 0, stream>>>(
            (float4*)acc, (const float4*)hn, total4);
        float* t = h; h = hn; hn = t;
    }

    // out[b] = dot(acc[users[b]], acc[items[b]+n_users]) / (n_layers+1)^2
    const float scale = 1.0f / ((float)(n_layers + 1) * (float)(n_layers + 1));
    const int waves = batch / 16;                 // 1024 tiles of 16
    const int score_threads = waves * 32;         // 32768
    lgcn_score_wmma_kernel<<<(score_threads + 255) / 256, 256, 0, stream>>>(
        acc, users, items, out, n_users, batch, scale);
}